// _ROIPool_19868518711718
// MI455X (gfx1250) — compile-verified
//
#include <hip/hip_runtime.h>

// ROI-Align for MI455X (gfx1250, wave32).
// Separable formulation: Out[c,py,px] = invA * sum_{h,w} Ay[py,h]*Ax[px,w]*X[c,h,w]
// Stage 1: y-contraction (coalesced streaming loads, VALU FMA).
// Stage 2: x-contraction via V_WMMA_F32_16X16X4_F32 (full f32 precision).
// v2: B fragments hoisted to registers (ROI-constant), double-buffered U ->
//     one barrier per channel tile instead of two, halved stage-2 DS traffic.

#define RP    7      // P_OUT
#define GG    4      // MAX_GRID
#define CTOT  256
#define HF    64
#define WF    64
#define KSPAN 32     // box span at scale 1/8 is <= 30 feature px -> 32 covers it
#define CTILE 16     // channels per WMMA tile (M dimension)

typedef float v2f __attribute__((ext_vector_type(2)));
typedef float v8f __attribute__((ext_vector_type(8)));

__global__ __launch_bounds__(256) void roi_align_wmma_kernel(
    const float* __restrict__ in,     // (B, 256, 64, 64)
    const float* __restrict__ boxes,  // (R, 4)
    const int*   __restrict__ bidx,   // (R,)
    float*       __restrict__ out)    // (R, 256, 7, 7)
{
    __shared__ float ay[RP][KSPAN];            // y interpolation+pool matrix
    __shared__ float ax[RP][KSPAN];            // x interpolation+pool matrix
    __shared__ float u[2][RP][KSPAN][CTILE];   // double-buffered U[py][w'][c]
    __shared__ float sc[8];                    // y1,x1,bin_h,bin_w,gh,gw,invA
    __shared__ int   si[4];                    // h0, w0, batch base

    const int r   = blockIdx.x;
    const int tid = threadIdx.x;

    // ---- Phase 0: per-ROI scalars (thread 0) ----
    if (tid == 0) {
        const float scale = (float)WF / 512.0f;   // 0.125
        float x1 = boxes[r*4+0] * scale;
        float y1 = boxes[r*4+1] * scale;
        float x2 = boxes[r*4+2] * scale;
        float y2 = boxes[r*4+3] * scale;
        float roi_w = fmaxf(x2 - x1, 1.0f);
        float roi_h = fmaxf(y2 - y1, 1.0f);
        float bin_w = roi_w / (float)RP;
        float bin_h = roi_h / (float)RP;
        float gh = fminf(fmaxf(ceilf(roi_h / (float)RP), 1.0f), (float)GG);
        float gw = fminf(fmaxf(ceilf(roi_w / (float)RP), 1.0f), (float)GG);
        // smallest valid sample coordinate (p=0, g=0) -> window origin
        float ymin = y1 + (0.5f / gh) * bin_h;
        float xmin = x1 + (0.5f / gw) * bin_w;
        si[0] = (int)floorf(fminf(fmaxf(ymin, 0.0f), (float)(HF - 1)));
        si[1] = (int)floorf(fminf(fmaxf(xmin, 0.0f), (float)(WF - 1)));
        si[2] = bidx[r] * CTOT * HF * WF;
        sc[0] = y1; sc[1] = x1; sc[2] = bin_h; sc[3] = bin_w;
        sc[4] = gh; sc[5] = gw; sc[6] = 1.0f / (gh * gw);
    }
    __syncthreads();

    const int h0 = si[0];
    const int w0 = si[1];
    const int base = si[2];
    const float inv_area = sc[6];

    // ---- Phase 1: build Ay (threads 0..6) and Ax (threads 32..38) rows ----
    // One thread per row: serial accumulation -> bitwise deterministic.
    if (tid < RP) {
        const float y1 = sc[0], bin_h = sc[2], gh = sc[4];
        float* row = ay[tid];
        #pragma unroll
        for (int k = 0; k < KSPAN; ++k) row[k] = 0.0f;
        for (int g = 0; g < GG; ++g) {
            if ((float)g < gh) {                    // grid-validity mask my
                float yy = y1 + ((float)tid + ((float)g + 0.5f) / gh) * bin_h;
                float valid = (yy >= -1.0f && yy <= (float)HF) ? 1.0f : 0.0f;
                float cc = fminf(fmaxf(yy, 0.0f), (float)(HF - 1));
                int lo = (int)floorf(cc);
                int hi = min(lo + 1, HF - 1);
                float fr = cc - (float)lo;
                int k1 = min(max(lo - h0, 0), KSPAN - 1);
                int k2 = min(max(hi - h0, 0), KSPAN - 1);
                row[k1] += (1.0f - fr) * valid;
                row[k2] += fr * valid;
            }
        }
    } else if (tid >= 32 && tid < 32 + RP) {
        const int px = tid - 32;
        const float x1 = sc[1], bin_w = sc[3], gw = sc[5];
        float* row = ax[px];
        #pragma unroll
        for (int k = 0; k < KSPAN; ++k) row[k] = 0.0f;
        for (int g = 0; g < GG; ++g) {
            if ((float)g < gw) {                    // grid-validity mask mx
                float xx = x1 + ((float)px + ((float)g + 0.5f) / gw) * bin_w;
                float valid = (xx >= -1.0f && xx <= (float)WF) ? 1.0f : 0.0f;
                float cc = fminf(fmaxf(xx, 0.0f), (float)(WF - 1));
                int lo = (int)floorf(cc);
                int hi = min(lo + 1, WF - 1);
                float fr = cc - (float)lo;
                int k1 = min(max(lo - w0, 0), KSPAN - 1);
                int k2 = min(max(hi - w0, 0), KSPAN - 1);
                row[k1] += (1.0f - fr) * valid;
                row[k2] += fr * valid;
            }
        }
    }
    __syncthreads();

    const int lane = tid & 31;
    const int wave = tid >> 5;
    const int lh   = lane >> 4;        // half-wave (0/1)
    const int n    = lane & 15;        // WMMA: A row m == B/D column n
    const int py   = (wave < 7) ? wave : 6;   // wave 7 duplicates py=6 (EXEC stays full), no store
    const int wcol = lane;             // stage-1 w' (coalesced: lane == consecutive w')
    const int widx = min(w0 + wcol, WF - 1);

    // ---- Hoist B fragments (Ax is ROI-constant across all channel tiles) ----
    // B 4x16 f32 layout: v0 = row (2*lh), v1 = row (2*lh+1), column = lane&15.
    v2f bfrag[KSPAN / 4];
    #pragma unroll
    for (int kc = 0; kc < KSPAN / 4; ++kc) {
        const int kb = kc * 4 + lh * 2;
        bfrag[kc].x = ax[n][kb];        // B[k=kb,   n] = Ax[px=n][w'=kb]
        bfrag[kc].y = ax[n][kb + 1];    // B[k=kb+1, n]
    }

    // ---- Main loop over 16 channel tiles (double-buffered U, 1 barrier/tile) ----
    for (int ct = 0; ct < CTOT / CTILE; ++ct) {
        const int c0  = ct * CTILE;
        const int buf = ct & 1;

        // Stage 1: U[py][w'][c] = sum_k Ay[py][k] * X[c][h0+k][w0+w']
        // thread -> (c = wave, wave+8 ; w' = lane); each wave streams 128B rows.
        #pragma unroll
        for (int cc = 0; cc < 2; ++cc) {
            const int cloc = wave + cc * 8;
            const float* src = in + base + (size_t)(c0 + cloc) * HF * WF;
            // prefetch the same rows of the next channel tile
            if (ct + 1 < CTOT / CTILE)
                __builtin_prefetch(src + CTILE * HF * WF + min(h0, HF-1) * WF + widx, 0, 1);
            float acc[RP];
            #pragma unroll
            for (int p = 0; p < RP; ++p) acc[p] = 0.0f;
            #pragma unroll 4
            for (int k = 0; k < KSPAN; ++k) {
                const int hidx = min(h0 + k, HF - 1);   // clamped; Ay==0 past span
                const float xv = src[hidx * WF + widx];
                #pragma unroll
                for (int p = 0; p < RP; ++p) acc[p] = fmaf(ay[p][k], xv, acc[p]);
            }
            #pragma unroll
            for (int p = 0; p < RP; ++p) u[buf][p][wcol][cloc] = acc[p];
        }
        // One barrier: orders this tile's U writes before its reads. Buffer
        // reuse at ct+2 is transitively protected by the barrier at ct+1.
        __syncthreads();

        // Stage 2: D[c, px] = sum_{w'} U[py][w'][c] * Ax[px][w']  via WMMA f32 16x16x4.
        // A 16x4 f32 layout: lanes 0-15 hold K=0(v0),K=1(v1); lanes 16-31 hold K=2,K=3.
        v8f d = {};
        #pragma unroll
        for (int kc = 0; kc < KSPAN / 4; ++kc) {
            const int kb = kc * 4 + lh * 2;
            v2f a;
            a.x = u[buf][py][kb][n];        // A[m=n, k=kb]
            a.y = u[buf][py][kb + 1][n];    // A[m=n, k=kb+1]
            d = __builtin_amdgcn_wmma_f32_16x16x4_f32(
                    false, a, false, bfrag[kc], (short)0, d, false, false);
        }

        // D layout: VGPR v holds rows m = v + 8*lh, column n = lane&15.
        if (wave < 7 && n < RP) {
            float* op = out + ((size_t)r * CTOT + c0 + 8 * lh) * (RP * RP)
                            + py * RP + n;
            #pragma unroll
            for (int v = 0; v < 8; ++v)
                op[(size_t)v * (RP * RP)] = d[v] * inv_area;
        }
    }
}

extern "C" void kernel_launch(void* const* d_in, const int* in_sizes, int n_in,
                              void* d_out, int out_size, void* d_ws, size_t ws_size,
                              hipStream_t stream) {
    const float* inputs = (const float*)d_in[0];
    const float* boxes  = (const float*)d_in[1];
    const int*   bidx   = (const int*)d_in[2];
    float* out          = (float*)d_out;
    const int R = in_sizes[1] / 4;   // boxes is (R,4)
    roi_align_wmma_kernel<<<R, 256, 0, stream>>>(inputs, boxes, bidx, out);
}